// OutputBlock_2439541424490
// MI455X (gfx1250) — compile-verified
//
#include <hip/hip_runtime.h>

// MI455X / gfx1250, wave32. fp32 end-to-end (matches reference numerics).
// Pipeline: zero h -> edge gate+scatter (bandwidth bound, ~350MB traffic)
//           -> WMMA f32 16x16x4 MLP head over [20000,128].

typedef __attribute__((ext_vector_type(2))) float v2f;
typedef __attribute__((ext_vector_type(8))) float v8f;

#define EMB 128
#define NRAD 6
#define NTGT 12
#define SW_PITCH 130   // weight LDS pitch: breaks half-wave bank aliasing (130 mod 64 = 2)
#define ACT_PITCH 132  // activation LDS pitch: conflict-free b64 A-frag reads (132 mod 64 = 4)

__device__ __forceinline__ v8f wmma4(v2f a, v2f b, v8f c) {
  // D = A(16x4 f32) * B(4x16 f32) + C(16x16 f32)
  return __builtin_amdgcn_wmma_f32_16x16x4_f32(false, a, false, b, (short)0, c,
                                               false, false);
}

__global__ void zero_f32x4(float4* __restrict__ p, int n4) {
  int i = blockIdx.x * blockDim.x + threadIdx.x;
  if (i < n4) p[i] = make_float4(0.f, 0.f, 0.f, 0.f);
}

// One wave32 per edge. Lane covers 4 consecutive channels (32*4 = 128).
__global__ __launch_bounds__(256)
void edge_gate_scatter(const float* __restrict__ x, const float* __restrict__ rbf,
                       const int* __restrict__ idnb, const float* __restrict__ Wrbf,
                       float* __restrict__ h, int nEdges) {
  __shared__ __align__(16) float sW[NRAD * EMB];  // 3 KB, stays resident
  for (int i = threadIdx.x; i < NRAD * EMB; i += blockDim.x) sW[i] = Wrbf[i];
  __syncthreads();

  const int lane = threadIdx.x & 31;
  int e = blockIdx.x * (blockDim.x >> 5) + (threadIdx.x >> 5);
  if (e >= nEdges) return;               // wave-uniform exit
  e = __builtin_amdgcn_readfirstlane(e); // force SGPR -> scalar loads for rbf/idx

  const float2 r01 = *(const float2*)(rbf + (size_t)e * NRAD + 0);
  const float2 r23 = *(const float2*)(rbf + (size_t)e * NRAD + 2);
  const float2 r45 = *(const float2*)(rbf + (size_t)e * NRAD + 4);
  const int atom = idnb[e];

  const int j = lane * 4;
  const float4 w0 = *(const float4*)(sW + 0 * EMB + j);
  const float4 w1 = *(const float4*)(sW + 1 * EMB + j);
  const float4 w2 = *(const float4*)(sW + 2 * EMB + j);
  const float4 w3 = *(const float4*)(sW + 3 * EMB + j);
  const float4 w4 = *(const float4*)(sW + 4 * EMB + j);
  const float4 w5 = *(const float4*)(sW + 5 * EMB + j);

  float4 g;
  g.x = r01.x*w0.x + r01.y*w1.x + r23.x*w2.x + r23.y*w3.x + r45.x*w4.x + r45.y*w5.x;
  g.y = r01.x*w0.y + r01.y*w1.y + r23.x*w2.y + r23.y*w3.y + r45.x*w4.y + r45.y*w5.y;
  g.z = r01.x*w0.z + r01.y*w1.z + r23.x*w2.z + r23.y*w3.z + r45.x*w4.z + r45.y*w5.z;
  g.w = r01.x*w0.w + r01.y*w1.w + r23.x*w2.w + r23.y*w3.w + r45.x*w4.w + r45.y*w5.w;

  const float4 xe = *(const float4*)(x + (size_t)e * EMB + j);  // global_load_b128
  float* dst = h + (size_t)atom * EMB + j;                      // L2-resident (10 MB)
  atomicAdd(dst + 0, g.x * xe.x);
  atomicAdd(dst + 1, g.y * xe.y);
  atomicAdd(dst + 2, g.z * xe.z);
  atomicAdd(dst + 3, g.w * xe.w);
}

// 256 threads = 8 waves; each wave owns 16 atom rows; block covers 128 rows.
__global__ __launch_bounds__(256)
void mlp_head(const float* __restrict__ hbuf, const float* __restrict__ Ws,
              const float* __restrict__ bs, const float* __restrict__ Wf,
              float* __restrict__ out, int nAtoms) {
  __shared__ __align__(16) float sW[EMB * SW_PITCH];       // 66.5 KB
  __shared__ __align__(16) float sB[EMB];                  // 0.5 KB
  __shared__ __align__(16) float sAct[8 * 16 * ACT_PITCH]; // 67.6 KB

  const int tid = threadIdx.x;
  const int lane = tid & 31;
  const int w = tid >> 5;
  const int hi = lane >> 4;  // 0: lanes 0-15, 1: lanes 16-31
  const int lo = lane & 15;
  const int rowBase = blockIdx.x * 128 + w * 16;
  float* act = sAct + w * (16 * ACT_PITCH);  // per-wave private tile

  // Stage h tile (16 rows x 128 cols) into LDS; OOB rows -> 0 (never stored).
  for (int m = 0; m < 16; ++m) {
    const int row = rowBase + m;
    float4 v = make_float4(0.f, 0.f, 0.f, 0.f);
    if (row < nAtoms) v = *(const float4*)(hbuf + (size_t)row * EMB + lane * 4);
    float* d = act + m * ACT_PITCH + lane * 4;
    d[0] = v.x; d[1] = v.y; d[2] = v.z; d[3] = v.w;
  }

  for (int layer = 0; layer < 3; ++layer) {
    __syncthreads();  // everyone done with previous sW before overwrite
    const float* Wsrc = Ws + layer * EMB * EMB;
    for (int i = tid; i < EMB * EMB / 2; i += blockDim.x) {
      const int idx = i * 2;
      const int k = idx >> 7, n = idx & 127;
      *(float2*)(&sW[k * SW_PITCH + n]) = *(const float2*)(Wsrc + idx);
    }
    if (tid < EMB / 4)
      *(float4*)(&sB[tid * 4]) = *(const float4*)(bs + layer * EMB + tid * 4);
    __syncthreads();

    // A-fragments: 16x4 f32 layout -> lane lo = row M, hi selects K-pair.
    v2f afrag[32];
#pragma unroll
    for (int kk = 0; kk < 32; ++kk)
      afrag[kk] = *(const v2f*)(act + lo * ACT_PITCH + kk * 4 + 2 * hi);

    v8f accs[8];
#pragma unroll
    for (int nn = 0; nn < 8; ++nn) {
      v8f acc = {};
#pragma unroll
      for (int kk = 0; kk < 32; ++kk) {
        const int k = kk * 4 + 2 * hi;
        v2f b;  // B 4x16: lane lo = col N, hi selects K-pair
        b.x = sW[k * SW_PITCH + nn * 16 + lo];
        b.y = sW[(k + 1) * SW_PITCH + nn * 16 + lo];
        acc = wmma4(afrag[kk], b, acc);
      }
      accs[nn] = acc;
    }

    // C layout: VGPR r -> row = r + 8*hi, col = nn*16 + lo. Bias + SiLU, back to LDS.
#pragma unroll
    for (int nn = 0; nn < 8; ++nn) {
#pragma unroll
      for (int r = 0; r < 8; ++r) {
        const int row = r + 8 * hi;
        const int col = nn * 16 + lo;
        const float vv = accs[nn][r] + sB[col];
        act[row * ACT_PITCH + col] = vv / (1.0f + __expf(-vv));
      }
    }
  }

  // Final 128 -> 12 projection (zero-padded to one 16-wide N-tile).
  __syncthreads();
  for (int i = tid; i < EMB * 16; i += blockDim.x) {
    const int k = i >> 4, n = i & 15;
    sW[i] = (n < NTGT) ? Wf[k * NTGT + n] : 0.0f;
  }
  __syncthreads();

  v8f acc = {};
#pragma unroll
  for (int kk = 0; kk < 32; ++kk) {
    const int k = kk * 4 + 2 * hi;
    const v2f a = *(const v2f*)(act + lo * ACT_PITCH + k);
    v2f b;
    b.x = sW[k * 16 + lo];
    b.y = sW[(k + 1) * 16 + lo];
    acc = wmma4(a, b, acc);
  }
#pragma unroll
  for (int r = 0; r < 8; ++r) {
    const int row = rowBase + r + 8 * hi;
    if (lo < NTGT && row < nAtoms) out[(size_t)row * NTGT + lo] = acc[r];
  }
}

extern "C" void kernel_launch(void* const* d_in, const int* in_sizes, int n_in,
                              void* d_out, int out_size, void* d_ws, size_t ws_size,
                              hipStream_t stream) {
  const float* x    = (const float*)d_in[0];  // [E,128]
  const float* rbf  = (const float*)d_in[1];  // [E,6]
  const int*   idnb = (const int*)d_in[2];    // [E]
  // d_in[3] = n_atoms (device scalar, unused host-side)
  const float* Wrbf = (const float*)d_in[4];  // [6,128]
  const float* Ws   = (const float*)d_in[5];  // [3,128,128]
  const float* bs   = (const float*)d_in[6];  // [3,128]
  const float* Wf   = (const float*)d_in[7];  // [128,12]
  float* out = (float*)d_out;

  const int nEdges = in_sizes[2];
  const int nAtoms = out_size / NTGT;
  float* h = (float*)d_ws;  // [nAtoms,128] f32 accumulator (10.24 MB)

  const int n4 = (nAtoms * EMB) / 4;
  zero_f32x4<<<(n4 + 255) / 256, 256, 0, stream>>>((float4*)h, n4);

  const int wavesPerBlock = 8;  // 256 threads
  const int eBlocks = (nEdges + wavesPerBlock - 1) / wavesPerBlock;
  edge_gate_scatter<<<eBlocks, 256, 0, stream>>>(x, rbf, idnb, Wrbf, h, nEdges);

  mlp_head<<<(nAtoms + 127) / 128, 256, 0, stream>>>(h, Ws, bs, Wf, out, nAtoms);
}